// Bias_8297876816109
// MI455X (gfx1250) — compile-verified
//
#include <hip/hip_runtime.h>
#include <hip/hip_bf16.h>

// Pure gather: out[i] = w[idx[i]], i in [0, 16384*512).
// Memory-bound: ~67 MB of stream traffic + 4 MB L2-resident gather table.
// Strategy: B128 NT loads/stores for the streams (keep L2 clean for w),
// default-RT scalar gathers from w (L2 hits), global_prefetch for the
// next index chunk. No matrix math exists in this op -> no WMMA by design.

typedef int   v4i __attribute__((ext_vector_type(4)));
typedef float v4f __attribute__((ext_vector_type(4)));

__global__ void __launch_bounds__(256)
gather_rows_kernel(const int* __restrict__ idx,
                   const float* __restrict__ w,
                   float* __restrict__ out,
                   long long n) {
    const long long n4     = n >> 2;                          // vec4 chunks
    const long long tid    = (long long)blockIdx.x * blockDim.x + threadIdx.x;
    const long long stride = (long long)gridDim.x * blockDim.x;

    const v4i* __restrict__ idx4 = (const v4i*)idx;
    v4f* __restrict__ out4       = (v4f*)out;

    for (long long i = tid; i < n4; i += stride) {
        // Prefetch next chunk of the index stream into cache hierarchy.
        // Speculative: past-the-end prefetch is silently dropped (ISA 10.5).
        __builtin_prefetch((const void*)(idx4 + i + stride), 0, 1);

        // Streaming read of 4 indices: one global_load_b128, NT temporal hint
        // so the index stream does not evict the hot 4 MB w table.
        v4i id = __builtin_nontemporal_load(idx4 + i);

        // 4 independent gathers from w (default RT policy -> stays L2-hot).
        v4f r;
        r.x = w[id.x];
        r.y = w[id.y];
        r.z = w[id.z];
        r.w = w[id.w];

        // Streaming write: one global_store_b128 with NT hint.
        __builtin_nontemporal_store(r, out4 + i);
    }

    // Scalar tail (n is divisible by 4 for this problem, but stay generic).
    for (long long i = (n4 << 2) + tid; i < n; i += stride) {
        out[i] = w[idx[i]];
    }
}

extern "C" void kernel_launch(void* const* d_in, const int* in_sizes, int n_in,
                              void* d_out, int out_size, void* d_ws, size_t ws_size,
                              hipStream_t stream) {
    const int*   idx = (const int*)d_in[0];   // input ids [16384, 512] (int32)
    const float* w   = (const float*)d_in[1]; // table [1'000'000, 1] (f32)
    float*       out = (float*)d_out;         // [16384, 512, 1] (f32)

    const long long n  = (long long)out_size;        // 8,388,608 elements
    const long long n4 = n >> 2;                     // 2,097,152 vec4 chunks

    const int block = 256;                           // 8 wave32s per block
    long long want  = (n4 + block - 1) / block;
    int grid = (int)(want < 4096 ? (want > 0 ? want : 1) : 4096); // ~2 iters/thread

    gather_rows_kernel<<<grid, block, 0, stream>>>(idx, w, out, n);
}